// BayesianBSpline1D_7395933684395
// MI455X (gfx1250) — compile-verified
//
#include <hip/hip_runtime.h>
#include <math.h>

typedef __attribute__((ext_vector_type(2))) float v2f;
typedef __attribute__((ext_vector_type(8))) float v8f;

#define N_COEFF   15
#define N_SAMPLES 8
#define ORDER     5
#define KH        0.2f     // knot spacing h = 2/N_BASIS
#define K0        (-2.0f)  // first knot = -1 - ORDER*h

// Branch-free tanh: tanh(x) = 1 - 2/(exp(2x)+1).
// Uses v_exp_f32 (via __expf) + v_rcp_f32; clamp keeps exp finite.
__device__ __forceinline__ float tanh_fast(float x) {
    const float xc = fminf(fmaxf(x, -9.0f), 9.0f);
    const float e  = __expf(2.0f * xc);
    const float r  = __builtin_amdgcn_rcpf(e + 1.0f);
    return fmaf(-2.0f, r, 1.0f);
}

// Dense Cox-de Boor with uniform knots; all knot constants fold at compile
// time (w1 = t - j/d, w2 = (j+d+1)/d - t). Produces b[0..14]; b[15] padded
// so the contraction is exactly K=16 for four accumulating 16x16x4 WMMAs.
__device__ __forceinline__ void bspline_basis(float xv, float b[20]) {
    const float u  = (xv - K0) * (1.0f / KH);   // u in [5,15] after tanh
    const float fj = floorf(u);
#pragma unroll
    for (int j = 0; j < 20; ++j)
        b[j] = (fj == (float)j) ? 1.0f : 0.0f;  // degree-0 basis
#pragma unroll
    for (int d = 1; d <= ORDER; ++d) {
        const float invd = 1.0f / ((float)d * KH);
        const float t = (xv - K0) * invd;
        const int m = 20 - d;
#pragma unroll
        for (int j = 0; j < m; ++j) {
            const float w1 = t - (float)j / (float)d;
            const float w2 = (float)(j + d + 1) / (float)d - t;
            b[j] = w1 * b[j] + w2 * b[j + 1];   // in-place ascending is safe
        }
    }
    b[15] = 0.0f;  // pad K dim 15 -> 16
}

__global__ __launch_bounds__(256)
void spline_kernel(const float* __restrict__ x,
                   const float* __restrict__ mu,
                   const float* __restrict__ lv,
                   const float* __restrict__ eps,
                   float* __restrict__ out, int n)
{
    __shared__ float s_cavg[16];
    const int lt = threadIdx.x;
    if (lt < 16) {
        float v = 0.0f;
        if (lt < N_COEFF) {
            float es = 0.0f;
#pragma unroll
            for (int s = 0; s < N_SAMPLES; ++s) es += eps[s * N_COEFF + lt];
            es *= (1.0f / (float)N_SAMPLES);
            v = mu[lt] + __expf(0.5f * lv[lt]) * es;  // c_avg[k]
        }
        s_cavg[lt] = v;
    }
    __syncthreads();

    const int lane = lt & 31;
    const bool lo = (lane < 16);

    // A operand: every row of A = c_avg slice (rows replicated).
    // 16x4 f32 A layout: lanes 0-15 -> K=4i,4i+1 ; lanes 16-31 -> K=4i+2,4i+3.
    v2f A[4];
#pragma unroll
    for (int i = 0; i < 4; ++i) {
        A[i].x = lo ? s_cavg[4 * i + 0] : s_cavg[4 * i + 2];
        A[i].y = lo ? s_cavg[4 * i + 1] : s_cavg[4 * i + 3];
    }

    const int tid      = blockIdx.x * blockDim.x + threadIdx.x;
    const int nthreads = gridDim.x * blockDim.x;
    const int wave_id  = tid >> 5;
    const int nwaves   = nthreads >> 5;
    const int tiles    = n >> 5;       // 32-point tiles (EXEC all-ones inside)

    for (int t = wave_id; t < tiles; t += nwaves) {
        const int p = (t << 5) + lane;
        const float xv = tanh_fast(x[p]);
        float b[20];
        bspline_basis(xv, b);

        v8f acc0 = {0.f, 0.f, 0.f, 0.f, 0.f, 0.f, 0.f, 0.f};
        v8f acc1 = {0.f, 0.f, 0.f, 0.f, 0.f, 0.f, 0.f, 0.f};
        const int src0 = lane & 15;        // sources: points t*32 + 0..15
        const int src1 = (lane & 15) + 16; // sources: points t*32 + 16..31

#pragma unroll
        for (int i = 0; i < 4; ++i) {
            const float e0 = b[4 * i + 0], e1 = b[4 * i + 1];
            const float e2 = b[4 * i + 2], e3 = b[4 * i + 3];
            {   // set 0: D columns = points of lanes 0-15
                const float t0 = __shfl(e0, src0, 32);
                const float t1 = __shfl(e1, src0, 32);
                const float t2 = __shfl(e2, src0, 32);
                const float t3 = __shfl(e3, src0, 32);
                v2f B; B.x = lo ? t0 : t2; B.y = lo ? t1 : t3;
                acc0 = __builtin_amdgcn_wmma_f32_16x16x4_f32(
                    false, A[i], false, B, (short)0, acc0, false, false);
            }
            {   // set 1: D columns = points of lanes 16-31
                const float t0 = __shfl(e0, src1, 32);
                const float t1 = __shfl(e1, src1, 32);
                const float t2 = __shfl(e2, src1, 32);
                const float t3 = __shfl(e3, src1, 32);
                v2f B; B.x = lo ? t0 : t2; B.y = lo ? t1 : t3;
                acc1 = __builtin_amdgcn_wmma_f32_16x16x4_f32(
                    false, A[i], false, B, (short)0, acc1, false, false);
            }
        }
        // All D rows identical (= per-point dots); C/D layout puts column
        // N = lane&15 in VGPR0 of each half => lane reads its own result.
        out[p] = lo ? acc0[0] : acc1[0];
    }

    // Tail (n not multiple of 32): scalar dot, no WMMA (EXEC may be partial).
    const int rem = tiles << 5;
    for (int p = rem + tid; p < n; p += nthreads) {
        const float xv = tanh_fast(x[p]);
        float b[20];
        bspline_basis(xv, b);
        float s = 0.0f;
#pragma unroll
        for (int k = 0; k < N_COEFF; ++k) s += b[k] * s_cavg[k];
        out[p] = s;
    }
}

__global__ void kl_kernel(const float* __restrict__ mu,
                          const float* __restrict__ lv,
                          float* __restrict__ out_kl)
{
    if (blockIdx.x == 0 && threadIdx.x == 0) {
        // log(var/25) = lv - log(25) exactly, since var = exp(lv).
        const float log25 = 3.2188758248682006f;
        float s = 0.0f;
#pragma unroll
        for (int k = 0; k < N_COEFF; ++k) {
            const float var = __expf(lv[k]);
            s += (var + mu[k] * mu[k]) * (1.0f / 25.0f) - 1.0f - lv[k] + log25;
        }
        out_kl[0] = 0.5f * s;
    }
}

extern "C" void kernel_launch(void* const* d_in, const int* in_sizes, int n_in,
                              void* d_out, int out_size, void* d_ws, size_t ws_size,
                              hipStream_t stream)
{
    const float* x  = (const float*)d_in[0];
    const float* mu = (const float*)d_in[1];
    const float* lv = (const float*)d_in[2];
    const float* ep = (const float*)d_in[3];
    float* out = (float*)d_out;
    const int n = in_sizes[0];

    int blocks = 4096;                      // ~1M threads, 4 points/thread at N=2^22
    const long long nt = (long long)blocks * 256;
    if ((long long)n < nt) blocks = (n + 255) / 256;
    if (blocks < 1) blocks = 1;

    spline_kernel<<<blocks, 256, 0, stream>>>(x, mu, lv, ep, out, n);
    kl_kernel<<<1, 32, 0, stream>>>(mu, lv, out + n);
}